// MultiChannelSelfAttention_57131654971560
// MI455X (gfx1250) — compile-verified
//
#include <hip/hip_runtime.h>

#define B_ 16
#define C_ 4
#define L_ 1024
#define D_ 256

#define NSEG 8
#define SEGLEN (L_ / NSEG)     // 128
#define XSTRIDE 260            // 256 + 4 pad floats: conflict-free ds_load_b64, keeps 16B align

typedef __attribute__((ext_vector_type(2))) float v2f;
typedef __attribute__((ext_vector_type(4))) float v4f;
typedef __attribute__((ext_vector_type(8))) float v8f;

// ---------------------------------------------------------------------------
// Kernel A: fused per-channel score  s = W2 . tanh(X W1 + b1) + b2
// One block = 16 rows of X (one WMMA M-tile) for one (b,c).
// X tile staged once in LDS (shared by all 4 waves); each wave owns 4 N-tiles.
// Uses V_WMMA_F32_16X16X4_F32, K-loop of 64 steps.
// ---------------------------------------------------------------------------
__device__ inline void accum_tile(const v8f& acc, int e, const float* __restrict__ b1c,
                                  const float* __restrict__ w2c, float hsum[8]) {
  const float b1v = b1c[e];
  const float w2v = w2c[e];
#pragma unroll
  for (int r = 0; r < 8; ++r) hsum[r] += w2v * tanhf(acc[r] + b1v);
}

__global__ __launch_bounds__(128) void score_kernel(
    const float* __restrict__ inp, const float* __restrict__ W1,
    const float* __restrict__ b1, const float* __restrict__ W2,
    const float* __restrict__ b2, float* __restrict__ s)
{
  __shared__ float ldsX[16 * XSTRIDE];   // 16.6 KB X tile (row-padded)
  __shared__ float lds_s[16];

  const int bc   = blockIdx.x >> 6;   // b*C + c
  const int mblk = blockIdx.x & 63;   // 16-row block within L
  const int c    = bc & (C_ - 1);
  const int lane = threadIdx.x & 31;
  const int wave = threadIdx.x >> 5;

  if (threadIdx.x < 16) lds_s[threadIdx.x] = 0.f;

  // Cooperative stage of the 16x256 X tile: 1024 v4f's over 128 threads.
  const float* xbase = inp + ((size_t)bc * L_ + (size_t)mblk * 16) * D_;
  for (int v = threadIdx.x; v < 16 * (D_ / 4); v += 128) {
    const int r  = v >> 6;          // row 0..15
    const int cq = v & 63;          // v4f index within row
    const v4f val = ((const v4f*)(xbase + (size_t)r * D_))[cq];
    *(v4f*)&ldsX[r * XSTRIDE + cq * 4] = val;
  }
  __syncthreads();

  const int   rowl  = lane & 15;
  const int   khalf = (lane >> 4) << 1;  // 0 or 2
  const float* W1c  = W1 + (size_t)c * D_ * D_;
  const int   col   = wave * 64 + (lane & 15);   // base e-column for this lane

  v8f acc0 = {}, acc1 = {}, acc2 = {}, acc3 = {};

  for (int k0 = 0; k0 < D_; k0 += 4) {
    const int kb = k0 + khalf;
    // A fragment from LDS (8B aligned, conflict-free with XSTRIDE pad)
    const v2f a = *(const v2f*)&ldsX[rowl * XSTRIDE + kb];

    const float* wk0 = W1c + (size_t)kb * D_;
    const float* wk1 = wk0 + D_;
    __builtin_prefetch(wk0 + 4 * D_ + col, 0, 3);   // next k-step of W1

    v2f bf;
    bf.x = wk0[col];      bf.y = wk1[col];
    acc0 = __builtin_amdgcn_wmma_f32_16x16x4_f32(false, a, false, bf, (short)0, acc0, false, false);
    bf.x = wk0[col + 16]; bf.y = wk1[col + 16];
    acc1 = __builtin_amdgcn_wmma_f32_16x16x4_f32(false, a, false, bf, (short)0, acc1, false, false);
    bf.x = wk0[col + 32]; bf.y = wk1[col + 32];
    acc2 = __builtin_amdgcn_wmma_f32_16x16x4_f32(false, a, false, bf, (short)0, acc2, false, false);
    bf.x = wk0[col + 48]; bf.y = wk1[col + 48];
    acc3 = __builtin_amdgcn_wmma_f32_16x16x4_f32(false, a, false, bf, (short)0, acc3, false, false);
  }

  // h = tanh(acc + b1); partial s contribution = h . W2 over this wave's columns
  const float* b1c = b1 + (size_t)c * D_;
  const float* w2c = W2 + (size_t)c * D_;
  float hsum[8];
#pragma unroll
  for (int r = 0; r < 8; ++r) hsum[r] = 0.f;
  accum_tile(acc0, col,      b1c, w2c, hsum);
  accum_tile(acc1, col + 16, b1c, w2c, hsum);
  accum_tile(acc2, col + 32, b1c, w2c, hsum);
  accum_tile(acc3, col + 48, b1c, w2c, hsum);

  // Reduce over the 16 N-lanes in each half-wave, then across waves via LDS.
  // C layout: VGPR r -> row M=r (lanes 0-15) / M=r+8 (lanes 16-31).
#pragma unroll
  for (int r = 0; r < 8; ++r) {
    float p = hsum[r];
    p += __shfl_xor(p, 1);
    p += __shfl_xor(p, 2);
    p += __shfl_xor(p, 4);
    p += __shfl_xor(p, 8);
    if ((lane & 15) == 0)
      atomicAdd(&lds_s[r + ((lane >> 4) << 3)], p);
  }
  __syncthreads();

  if (threadIdx.x < 16)
    s[(size_t)bc * L_ + mblk * 16 + threadIdx.x] = lds_s[threadIdx.x] + b2[c];
}

// ---------------------------------------------------------------------------
// Kernel B: per (b,c) row of s: global max, E = exp(s - m), inclusive prefix
// sum Z_i, store E and 1/Z_i. One 1024-thread block per (b,c).
// ---------------------------------------------------------------------------
__global__ __launch_bounds__(1024) void softmax_scan_kernel(
    const float* __restrict__ s, float* __restrict__ E, float* __restrict__ rZ)
{
  __shared__ float red[1024];
  const int bc = blockIdx.x;
  const int i  = threadIdx.x;

  const float sv = s[bc * L_ + i];
  red[i] = sv;
  __syncthreads();
  for (int off = 512; off > 0; off >>= 1) {
    if (i < off) red[i] = fmaxf(red[i], red[i + off]);
    __syncthreads();
  }
  const float gm = red[0];
  __syncthreads();

  const float e = __expf(sv - gm);
  E[bc * L_ + i] = e;

  red[i] = e;
  __syncthreads();
  for (int off = 1; off < 1024; off <<= 1) {
    const float t = (i >= off) ? red[i - off] : 0.f;
    __syncthreads();
    red[i] += t;
    __syncthreads();
  }
  rZ[bc * L_ + i] = 1.f / red[i];
}

// ---------------------------------------------------------------------------
// Kernel C: attn[b,c,i,j] = (j<=i) ? E[j]/Z_i : 0. Pure bandwidth (268 MB).
// One block per attention row; v4f nontemporal stores (write-once stream
// larger than the 192 MB L2 -> bypass with TH=NT).
// ---------------------------------------------------------------------------
__global__ __launch_bounds__(256) void attn_write_kernel(
    const float* __restrict__ E, const float* __restrict__ rZ,
    float* __restrict__ attn)
{
  const int bc = blockIdx.x >> 10;
  const int i  = blockIdx.x & 1023;
  const float rz = rZ[bc * L_ + i];
  const v4f ev = ((const v4f*)E)[bc * (L_ / 4) + threadIdx.x];
  const int j = threadIdx.x << 2;
  v4f o;
  o.x = (j + 0 <= i) ? ev.x * rz : 0.f;
  o.y = (j + 1 <= i) ? ev.y * rz : 0.f;
  o.z = (j + 2 <= i) ? ev.z * rz : 0.f;
  o.w = (j + 3 <= i) ? ev.w * rz : 0.f;
  __builtin_nontemporal_store(o, &((v4f*)attn)[((size_t)blockIdx.x << 8) + threadIdx.x]);
}

// ---------------------------------------------------------------------------
// Kernel D (pass 1): per-segment sums  S[bc][g][d] = sum_{i in seg g} E[i]*x[i,d]
// 8 segments of 128 rows -> 512 blocks for good WGP coverage.
// ---------------------------------------------------------------------------
__global__ __launch_bounds__(256) void seg_sum_kernel(
    const float* __restrict__ inp, const float* __restrict__ E,
    float* __restrict__ segS)
{
  const int g  = blockIdx.x & (NSEG - 1);
  const int bc = blockIdx.x >> 3;
  const int d  = threadIdx.x;
  const float* x  = inp + (size_t)bc * L_ * D_ + (size_t)g * SEGLEN * D_;
  const float* Eb = E + bc * L_ + g * SEGLEN;

  float acc = 0.f;
  for (int i = 0; i < SEGLEN; ++i)
    acc = fmaf(Eb[i], x[(size_t)i * D_ + d], acc);
  segS[((size_t)bc * NSEG + g) * D_ + d] = acc;
}

// ---------------------------------------------------------------------------
// Kernel D (pass 2): out[i,d] = (prefix_of_earlier_segments + running scan)/Z_i
// ---------------------------------------------------------------------------
__global__ __launch_bounds__(256) void out_seg_scan_kernel(
    const float* __restrict__ inp, const float* __restrict__ E,
    const float* __restrict__ rZ, const float* __restrict__ segS,
    float* __restrict__ out)
{
  const int g  = blockIdx.x & (NSEG - 1);
  const int bc = blockIdx.x >> 3;
  const int d  = threadIdx.x;
  const float* x   = inp + (size_t)bc * L_ * D_ + (size_t)g * SEGLEN * D_;
  float*       o   = out + (size_t)bc * L_ * D_ + (size_t)g * SEGLEN * D_;
  const float* Eb  = E  + bc * L_ + g * SEGLEN;
  const float* rZb = rZ + bc * L_ + g * SEGLEN;

  float acc = 0.f;
  for (int gp = 0; gp < g; ++gp)
    acc += segS[((size_t)bc * NSEG + gp) * D_ + d];

  for (int i = 0; i < SEGLEN; ++i) {
    acc = fmaf(Eb[i], x[(size_t)i * D_ + d], acc);
    __builtin_nontemporal_store(acc * rZb[i], &o[(size_t)i * D_ + d]);
  }
}

// ---------------------------------------------------------------------------
extern "C" void kernel_launch(void* const* d_in, const int* in_sizes, int n_in,
                              void* d_out, int out_size, void* d_ws, size_t ws_size,
                              hipStream_t stream)
{
  const float* inp = (const float*)d_in[0];
  // d_in[1] = mask (bool) — causality is handled analytically, unused.
  const float* W1 = (const float*)d_in[2];
  const float* b1 = (const float*)d_in[3];
  const float* W2 = (const float*)d_in[4];
  const float* b2 = (const float*)d_in[5];

  float* out  = (float*)d_out;                            // [B,C,L,D]
  float* attn = out + (size_t)B_ * C_ * L_ * D_;          // [B,C,L,L]

  float* s    = (float*)d_ws;                             // [B*C*L]
  float* E    = s + (size_t)B_ * C_ * L_;                 // [B*C*L]
  float* rZ   = E + (size_t)B_ * C_ * L_;                 // [B*C*L]
  float* segS = rZ + (size_t)B_ * C_ * L_;                // [B*C*NSEG*D]

  score_kernel<<<B_ * C_ * (L_ / 16), 128, 0, stream>>>(inp, W1, b1, W2, b2, s);
  softmax_scan_kernel<<<B_ * C_, 1024, 0, stream>>>(s, E, rZ);
  attn_write_kernel<<<B_ * C_ * L_, 256, 0, stream>>>(E, rZ, attn);
  seg_sum_kernel<<<B_ * C_ * NSEG, 256, 0, stream>>>(inp, E, segS);
  out_seg_scan_kernel<<<B_ * C_ * NSEG, 256, 0, stream>>>(inp, E, rZ, segS, out);
}